// CustomModel_lipsnet_83296595739102
// MI455X (gfx1250) — compile-verified
//
#include <hip/hip_runtime.h>
#include <math.h>

typedef __attribute__((ext_vector_type(2))) float v2f;
typedef __attribute__((ext_vector_type(8))) float v8f;

#define NP   96    // COMP padded to multiple of 16
#define COMP 85
#define CON  256
#define HID  128
#define NOBS 64
#define NACT 16
#define NKS  32
#define LIPS_EPS 1e-4f

// Shared memory layout (floats): 3 big 96x96 tiles + small scratch
#define SMEM_FLOATS (3*NP*NP + 256 + 128 + 80 + 6*NP + 128 + 4)

// -------------------------------------------------------------------------
// Kernel 1: G1 = mw1 @ mw1^T, zero-padded to 96x96. Sample-independent.
// -------------------------------------------------------------------------
__global__ __launch_bounds__(NP) void lips_g1_kernel(const float* __restrict__ mw1,
                                                     float* __restrict__ g1) {
  __shared__ float row[CON];
  const int r = blockIdx.x;
  const int c = threadIdx.x;
  for (int k = c; k < CON; k += NP) row[k] = (r < COMP) ? mw1[(size_t)r * CON + k] : 0.f;
  __syncthreads();
  float s = 0.f;
  if (r < COMP && c < COMP) {
    const float* wr = mw1 + (size_t)c * CON;
#pragma unroll 8
    for (int k = 0; k < CON; ++k) s = fmaf(row[k], wr[k], s);
  }
  g1[r * NP + c] = s;
}

// -------------------------------------------------------------------------
// Kernel 2: fully-fused per-sample LipsNet. 1 workgroup (4 wave32) / sample.
// ||J||^2 = tr(P G1 P^T) = sum(P .* (P@G1)),  P = (D3 W3)(D2 W2 D1)
// -------------------------------------------------------------------------
__global__ __launch_bounds__(128) void lips_main_kernel(
    const float* __restrict__ obs, const float* __restrict__ act,
    const float* __restrict__ ow1, const float* __restrict__ ob1,
    const float* __restrict__ ow2, const float* __restrict__ ob2,
    const float* __restrict__ aw1, const float* __restrict__ ab1,
    const float* __restrict__ aw2, const float* __restrict__ ab2,
    const float* __restrict__ mw1, const float* __restrict__ mb1,
    const float* __restrict__ mw2, const float* __restrict__ mb2,
    const float* __restrict__ mw3, const float* __restrict__ mb3,
    const float* __restrict__ kw1, const float* __restrict__ kb1,
    const float* __restrict__ kw2, const float* __restrict__ kb2,
    const float* __restrict__ kw3, const float* __restrict__ kb3,
    const float* __restrict__ g1, float* __restrict__ out)
{
  extern __shared__ float smem[];
  float* sA   = smem;              // 96x96 : masked W3' (then reused for G1)
  float* sC   = sA + NP * NP;      // 96x96 : masked W2 (D2 W2 D1)
  float* sP   = sC + NP * NP;      // 96x96 : P
  float* sX   = sP + NP * NP;      // 256   : concat features
  float* sT   = sX + 256;          // 128   : hidden scratch / k-net
  float* sIn  = sT + 128;          // 80    : obs row (64) + act row (16)
  float* sH1  = sIn + 80;          // 96
  float* sH2  = sH1 + NP;          // 96
  float* sD1  = sH2 + NP;          // 96
  float* sD2  = sD1 + NP;          // 96
  float* sD3  = sD2 + NP;          // 96
  float* sF3  = sD3 + NP;          // 96
  float* sRed = sF3 + NP;          // 128
  float* sS   = sRed + 128;        // [0]=k_out, [1]=jac_norm

  const int tid = threadIdx.x;
  const int b   = blockIdx.x;

  // ---- stage input rows ----
  if (tid < NOBS)             sIn[tid] = obs[(size_t)b * NOBS + tid];
  else if (tid < NOBS + NACT) sIn[tid] = act[(size_t)b * NACT + (tid - NOBS)];
  __syncthreads();

  // ---- obs head, layer 1: 64 -> 128 ----
  {
    float s = ob1[tid];
    const float* w = ow1 + (size_t)tid * NOBS;
#pragma unroll 8
    for (int k = 0; k < NOBS; ++k) s = fmaf(w[k], sIn[k], s);
    sT[tid] = fmaxf(s, 0.f);
  }
  __syncthreads();
  // ---- obs head, layer 2: 128 -> 128 ----
  {
    float s = ob2[tid];
    const float* w = ow2 + (size_t)tid * HID;
#pragma unroll 8
    for (int k = 0; k < HID; ++k) s = fmaf(w[k], sT[k], s);
    sX[tid] = fmaxf(s, 0.f);
  }
  __syncthreads();
  // ---- action head, layer 1: 16 -> 128 ----
  {
    float s = ab1[tid];
    const float* w = aw1 + (size_t)tid * NACT;
#pragma unroll
    for (int k = 0; k < NACT; ++k) s = fmaf(w[k], sIn[NOBS + k], s);
    sT[tid] = fmaxf(s, 0.f);
  }
  __syncthreads();
  // ---- action head, layer 2: 128 -> 128 ----
  {
    float s = ab2[tid];
    const float* w = aw2 + (size_t)tid * HID;
#pragma unroll 8
    for (int k = 0; k < HID; ++k) s = fmaf(w[k], sT[k], s);
    sX[HID + tid] = fmaxf(s, 0.f);
  }
  __syncthreads();

  // ---- k-net: 256 -> 32 (tanh) ----
  if (tid < NKS) {
    float s = kb1[tid];
    const float* w = kw1 + (size_t)tid * CON;
#pragma unroll 8
    for (int k = 0; k < CON; ++k) s = fmaf(w[k], sX[k], s);
    sT[tid] = tanhf(s);
  }
  __syncthreads();
  // ---- k-net: 32 -> 16 (tanh) ----
  if (tid < 16) {
    float s = kb2[tid];
    const float* w = kw2 + (size_t)tid * NKS;
#pragma unroll
    for (int k = 0; k < NKS; ++k) s = fmaf(w[k], sT[k], s);
    sT[32 + tid] = tanhf(s);
  }
  __syncthreads();
  // ---- k-net: 16 -> 1 (softplus) ----
  if (tid == 0) {
    float s = kb3[0];
#pragma unroll
    for (int k = 0; k < 16; ++k) s = fmaf(kw3[k], sT[32 + k], s);
    sS[0] = (s > 20.f) ? s : log1pf(expf(s));
  }
  // ---- f-net layer 1: 256 -> 85 (keep mask) ----
  if (tid < NP) {
    float p = 0.f;
    if (tid < COMP) {
      p = mb1[tid];
      const float* w = mw1 + (size_t)tid * CON;
#pragma unroll 8
      for (int k = 0; k < CON; ++k) p = fmaf(w[k], sX[k], p);
    }
    sH1[tid] = fmaxf(p, 0.f);
    sD1[tid] = (tid < COMP && p > 0.f) ? 1.f : 0.f;
  }
  __syncthreads();
  // ---- f-net layer 2 ----
  if (tid < NP) {
    float p = 0.f;
    if (tid < COMP) {
      p = mb2[tid];
      const float* w = mw2 + (size_t)tid * COMP;
#pragma unroll 5
      for (int k = 0; k < COMP; ++k) p = fmaf(w[k], sH1[k], p);
    }
    sH2[tid] = fmaxf(p, 0.f);
    sD2[tid] = (tid < COMP && p > 0.f) ? 1.f : 0.f;
  }
  __syncthreads();
  // ---- f-net layer 3 ----
  if (tid < NP) {
    float p = 0.f;
    if (tid < COMP) {
      p = mb3[tid];
      const float* w = mw3 + (size_t)tid * COMP;
#pragma unroll 5
      for (int k = 0; k < COMP; ++k) p = fmaf(w[k], sH2[k], p);
    }
    sF3[tid] = fmaxf(p, 0.f);
    sD3[tid] = (tid < COMP && p > 0.f) ? 1.f : 0.f;
  }
  __syncthreads();

  // ---- stage masked, zero-padded W3' = D3*W3 and C = D2*W2*D1 into LDS ----
  for (int idx = tid; idx < NP * NP; idx += 128) {
    const int r = idx / NP;
    const int c = idx - r * NP;
    float w3 = 0.f, w2 = 0.f;
    if (r < COMP && c < COMP) {
      w3 = sD3[r] * mw3[r * COMP + c];
      w2 = sD2[r] * mw2[r * COMP + c] * sD1[c];
    }
    sA[idx] = w3;
    sC[idx] = w2;
  }
  __syncthreads();

  // ---- WMMA setup (wave32): fragment lane mapping per CDNA5 ISA ----
  const int lane = tid & 31;
  const int wv   = tid >> 5;       // 4 waves, 9 tiles each (6x6 grid)
  const int half = lane >> 4;      // K-pair select for A/B frags
  const int mr   = lane & 15;

  // ---- P = W3' @ C  (96x96x96, fp32 WMMA) ----
  for (int tt = wv * 9; tt < wv * 9 + 9; ++tt) {
    const int tm = tt / 6, tn = tt - tm * 6;
    const int arow = tm * 16 + mr;
    const int bcol = tn * 16 + mr;
    v8f acc = {};
#pragma unroll
    for (int kk = 0; kk < NP; kk += 4) {
      const int k0 = kk + 2 * half;
      v2f a = *(const v2f*)(sA + arow * NP + k0);   // A[m][k0], A[m][k0+1]
      v2f bb;
      bb.x = sC[k0 * NP + bcol];                     // B[k0][n]
      bb.y = sC[(k0 + 1) * NP + bcol];               // B[k0+1][n]
      acc = __builtin_amdgcn_wmma_f32_16x16x4_f32(false, a, false, bb,
                                                  (short)0, acc, false, false);
    }
    const int prow = tm * 16 + 8 * half;
#pragma unroll
    for (int v = 0; v < 8; ++v) sP[(prow + v) * NP + bcol] = acc[v];
  }
  __syncthreads();

  // ---- reuse sA buffer: load padded G1 ----
  for (int idx = tid; idx < NP * NP; idx += 128) sA[idx] = g1[idx];
  __syncthreads();

  // ---- Q = P @ G1, fold sum(P .* Q) directly ----
  float part = 0.f;
  for (int tt = wv * 9; tt < wv * 9 + 9; ++tt) {
    const int tm = tt / 6, tn = tt - tm * 6;
    const int arow = tm * 16 + mr;
    const int bcol = tn * 16 + mr;
    v8f acc = {};
#pragma unroll
    for (int kk = 0; kk < NP; kk += 4) {
      const int k0 = kk + 2 * half;
      v2f a = *(const v2f*)(sP + arow * NP + k0);
      v2f bb;
      bb.x = sA[k0 * NP + bcol];
      bb.y = sA[(k0 + 1) * NP + bcol];
      acc = __builtin_amdgcn_wmma_f32_16x16x4_f32(false, a, false, bb,
                                                  (short)0, acc, false, false);
    }
    const int prow = tm * 16 + 8 * half;
#pragma unroll
    for (int v = 0; v < 8; ++v) part = fmaf(acc[v], sP[(prow + v) * NP + bcol], part);
  }
  sRed[tid] = part;
  __syncthreads();
#pragma unroll
  for (int off = 64; off > 0; off >>= 1) {
    if (tid < off) sRed[tid] += sRed[tid + off];
    __syncthreads();
  }
  if (tid == 0) sS[1] = sqrtf(sRed[0]);
  __syncthreads();

  // ---- output: tanh(k * f_out / (||J|| + eps)) ----
  if (tid < COMP) {
    const float v = sS[0] * sF3[tid] / (sS[1] + LIPS_EPS);
    out[(size_t)b * COMP + tid] = tanhf(v);
  }
}

// -------------------------------------------------------------------------
extern "C" void kernel_launch(void* const* d_in, const int* in_sizes, int n_in,
                              void* d_out, int out_size, void* d_ws, size_t ws_size,
                              hipStream_t stream) {
  (void)n_in; (void)out_size; (void)ws_size;
  const float* obs = (const float*)d_in[0];
  const float* act = (const float*)d_in[1];
  const float* ow1 = (const float*)d_in[2];
  const float* ob1 = (const float*)d_in[3];
  const float* ow2 = (const float*)d_in[4];
  const float* ob2 = (const float*)d_in[5];
  const float* aw1 = (const float*)d_in[6];
  const float* ab1 = (const float*)d_in[7];
  const float* aw2 = (const float*)d_in[8];
  const float* ab2 = (const float*)d_in[9];
  const float* mw1 = (const float*)d_in[10];
  const float* mb1 = (const float*)d_in[11];
  const float* mw2 = (const float*)d_in[12];
  const float* mb2 = (const float*)d_in[13];
  const float* mw3 = (const float*)d_in[14];
  const float* mb3 = (const float*)d_in[15];
  const float* kw1 = (const float*)d_in[16];
  const float* kb1 = (const float*)d_in[17];
  const float* kw2 = (const float*)d_in[18];
  const float* kb2 = (const float*)d_in[19];
  const float* kw3 = (const float*)d_in[20];
  const float* kb3 = (const float*)d_in[21];

  float* g1 = (float*)d_ws;                 // 96*96 floats = 36,864 B
  const int B = in_sizes[0] / NOBS;         // 8192

  lips_g1_kernel<<<NP, NP, 0, stream>>>(mw1, g1);

  const size_t smem = (size_t)SMEM_FLOATS * sizeof(float);
  lips_main_kernel<<<B, 128, smem, stream>>>(
      obs, act, ow1, ob1, ow2, ob2, aw1, ab1, aw2, ab2,
      mw1, mb1, mw2, mb2, mw3, mb3, kw1, kb1, kw2, kb2, kw3, kb3,
      g1, (float*)d_out);
}